// FrustumFeatureEncoder_12360915878212
// MI455X (gfx1250) — compile-verified
//
#include <hip/hip_runtime.h>
#include <math.h>

typedef __attribute__((ext_vector_type(2))) float v2f;
typedef __attribute__((ext_vector_type(8))) float v8f;

#define NPTS   2000000
#define NAZ    512
#define NKEY   32768            // 64 * 512
#define EPSBN  1e-5f

// ---- workspace layout (bytes) ----
#define RANK_OFF  0u            // uint[32768]  : occupancy flag -> voxel rank
#define VACC_OFF  131072u       // float4[32768]: sum x,y,z,count -> mean x,y,z,count
#define STATS_OFF 655360u       // double[44]   : 8 sums + 36 upper-tri 2nd moments
#define WF_OFF    656384u       // float[32*8]  : fully folded layer-0 weight
#define BF_OFF    657408u       // float[32]    : fully folded layer-0 bias
#define V_OFF     657536u       // uint         : number of occupied voxels
#define VMAXM_OFF 1048576u      // uint[32768*64]: monotone-munged segment max (8 MB, L2 resident)

__device__ __forceinline__ unsigned mungf(float f) {
  unsigned b = __float_as_uint(f);
  return (b & 0x80000000u) ? ~b : (b | 0x80000000u);
}
__device__ __forceinline__ float unmungf(unsigned k) {
  unsigned b = (k & 0x80000000u) ? (k ^ 0x80000000u) : ~k;
  return __uint_as_float(b);
}
__device__ __forceinline__ v8f wmma4(v2f a, v2f b, v8f c) {
  // V_WMMA_F32_16X16X4_F32 : D = A(16x4 f32) * B(4x16 f32) + C(16x16 f32)
  return __builtin_amdgcn_wmma_f32_16x16x4_f32(false, a, false, b, (short)0, c,
                                               false, false);
}

// ---------------- K1: init workspace ----------------
__global__ void ffe_init(unsigned char* ws) {
  size_t i = (size_t)blockIdx.x * 256 + threadIdx.x;   // grid covers NKEY*64 exactly
  ((unsigned*)(ws + VMAXM_OFF))[i] = 0x007FFFFFu;      // munged -inf
  if (i < NKEY) {
    ((unsigned*)(ws + RANK_OFF))[i] = 0u;
    ((float4*)(ws + VACC_OFF))[i] = make_float4(0.f, 0.f, 0.f, 0.f);
  }
  if (i < 44) ((double*)(ws + STATS_OFF))[i] = 0.0;
  if (i == 0) *(unsigned*)(ws + V_OFF) = 0u;
}

// ---------------- K2: occupancy + per-voxel feature sums ----------------
__global__ void ffe_scatter(const float* __restrict__ feat,
                            const int* __restrict__ coors, unsigned char* ws) {
  int i = blockIdx.x * blockDim.x + threadIdx.x;
  if (i >= NPTS) return;
  float4 f = ((const float4*)feat)[i];
  int4 cc = ((const int4*)coors)[i];
  unsigned key = (unsigned)cc.y * NAZ + (unsigned)cc.z;
  ((unsigned*)(ws + RANK_OFF))[key] = 1u;              // benign race, all write 1
  float* va = (float*)(ws + VACC_OFF) + (size_t)key * 4;
  atomicAdd(va + 0, f.x);
  atomicAdd(va + 1, f.y);
  atomicAdd(va + 2, f.z);
  atomicAdd(va + 3, 1.0f);
}

// ---------------- K3: single-block scan -> ranks, V, voxel_coors, means ----
__global__ void ffe_scan(float* __restrict__ out, unsigned char* ws) {
  __shared__ unsigned part[1024];
  unsigned* rank = (unsigned*)(ws + RANK_OFF);
  float4* vacc = (float4*)(ws + VACC_OFF);
  int t = threadIdx.x;
  unsigned flags[32], loc = 0;
#pragma unroll
  for (int i = 0; i < 32; ++i) { flags[i] = rank[t * 32 + i]; loc += flags[i]; }
  part[t] = loc;
  __syncthreads();
  for (int o = 1; o < 1024; o <<= 1) {           // Hillis-Steele inclusive scan
    unsigned v = (t >= o) ? part[t - o] : 0u;
    __syncthreads();
    part[t] += v;
    __syncthreads();
  }
  unsigned run = part[t] - loc;                  // exclusive prefix of this chunk
  unsigned Vtot = part[1023];
#pragma unroll 4
  for (int i = 0; i < 32; ++i) {
    unsigned key = t * 32 + i;
    if (flags[i]) {
      rank[key] = run;
      float4 a = vacc[key];
      float inv = 1.f / a.w;
      vacc[key] = make_float4(a.x * inv, a.y * inv, a.z * inv, a.w);
      float* co = out + (size_t)Vtot * 64 + (size_t)run * 4;  // voxel_coors block
      co[0] = 0.f; co[1] = (float)(key >> 9); co[2] = (float)(key & 511u); co[3] = 0.f;
      ++run;
    } else {
      rank[key] = run;
    }
  }
  if (t == 0) *(unsigned*)(ws + V_OFF) = Vtot;
}

// ---------------- K4: batch statistics of the 8-dim feature x -------------
__global__ void ffe_stats(const float* __restrict__ feat,
                          const int* __restrict__ coors, unsigned char* ws) {
  const float4* vacc = (const float4*)(ws + VACC_OFF);
  double* st = (double*)(ws + STATS_OFF);
  float s[8], mm[36];
#pragma unroll
  for (int i = 0; i < 8; ++i) s[i] = 0.f;
#pragma unroll
  for (int i = 0; i < 36; ++i) mm[i] = 0.f;
  unsigned stride = gridDim.x * blockDim.x;
  for (unsigned p = blockIdx.x * blockDim.x + threadIdx.x; p < NPTS; p += stride) {
    float4 f = ((const float4*)feat)[p];
    int4 cc = ((const int4*)coors)[p];
    unsigned key = (unsigned)cc.y * NAZ + (unsigned)cc.z;
    float4 vm = vacc[key];
    float dist = __builtin_amdgcn_sqrtf(f.x * f.x + f.y * f.y + f.z * f.z);
    float x8[8] = {f.x, f.y, f.z, f.w, dist, f.x - vm.x, f.y - vm.y, f.z - vm.z};
    int q = 0;
#pragma unroll
    for (int i = 0; i < 8; ++i) {
      s[i] += x8[i];
#pragma unroll
      for (int k = i; k < 8; ++k) mm[q++] += x8[i] * x8[k];
    }
  }
  __shared__ float red[256];
  int tid = threadIdx.x;
  for (int q = 0; q < 44; ++q) {
    red[tid] = (q < 8) ? s[q] : mm[q - 8];
    __syncthreads();
    for (int o = 128; o > 0; o >>= 1) {
      if (tid < o) red[tid] += red[tid + o];
      __syncthreads();
    }
    if (tid == 0) atomicAdd(&st[q], (double)red[0]);
    __syncthreads();
  }
}

// ---------------- K5: fold pre_norm + bn0 into Wf, bf ---------------------
__global__ void ffe_fold(const float* __restrict__ pre_g, const float* __restrict__ pre_b,
                         const float* __restrict__ W0, const float* __restrict__ g0,
                         const float* __restrict__ b0, unsigned char* ws) {
  int j = threadIdx.x;
  if (j >= 32) return;
  const double* st = (const double*)(ws + STATS_OFF);
  float m[8], C[8][8], sc[8], tt[8];
  const float invN = 1.0f / (float)NPTS;
#pragma unroll
  for (int i = 0; i < 8; ++i) m[i] = (float)(st[i] * (double)invN);
  int q = 8;
#pragma unroll
  for (int i = 0; i < 8; ++i)
#pragma unroll
    for (int k = i; k < 8; ++k) {
      float c = (float)(st[q++] * (double)invN) - m[i] * m[k];
      C[i][k] = c; C[k][i] = c;
    }
#pragma unroll
  for (int i = 0; i < 8; ++i) {
    sc[i] = pre_g[i] * rsqrtf(C[i][i] + EPSBN);          // xn = x*sc + tt
    tt[i] = pre_b[i] - m[i] * sc[i];
  }
  float vy = 0.f, my = 0.f, wt = 0.f;
#pragma unroll
  for (int i = 0; i < 8; ++i) {
    float wi = W0[j * 8 + i];
    my += wi * pre_b[i];                                  // E[y_j] (exact)
    wt += wi * tt[i];
#pragma unroll
    for (int k = 0; k < 8; ++k) vy += wi * sc[i] * C[i][k] * sc[k] * W0[j * 8 + k];
  }
  float rj = rsqrtf(vy + EPSBN);
  float gj = g0[j] * rj;
  float* Wf = (float*)(ws + WF_OFF);
  float* bf = (float*)(ws + BF_OFF);
#pragma unroll
  for (int i = 0; i < 8; ++i) Wf[j * 8 + i] = W0[j * 8 + i] * sc[i] * gj;
  bf[j] = b0[j] + gj * (wt - my);
}

// ---------------- K6: main WMMA pass (8->32 relu, 32->64, seg-max) --------
__global__ void __launch_bounds__(256) ffe_main(
    const float* __restrict__ feat, const int* __restrict__ coors,
    const float* __restrict__ W1, const float* __restrict__ b1,
    float* __restrict__ out, unsigned char* ws) {
  const unsigned* rank = (const unsigned*)(ws + RANK_OFF);
  const float4* vacc = (const float4*)(ws + VACC_OFF);
  const float* Wf = (const float*)(ws + WF_OFF);
  const float* bf = (const float*)(ws + BF_OFF);
  unsigned* vmaxm = (unsigned*)(ws + VMAXM_OFF);
  unsigned V = *(const unsigned*)(ws + V_OFF);
  float* f0o = out + (size_t)V * 68;                 // after voxel_feats + coors
  float* f1o = f0o + (size_t)NPTS * 32;

  __shared__ float w1lds[2048];                      // W1 staged via async DMA (8KB)
  __shared__ float ldsf0[8 * 16 * 36];               // per-wave 16x36 f0 tile
  __shared__ float ldsf1[8 * 16 * 68];               // per-wave 16x68 f1 tile
  __shared__ unsigned ldsv[8 * 16];                  // per-wave voxel ids per row
  const int wave = threadIdx.x >> 5;
  const int lane = threadIdx.x & 31;
  const int mrow = lane & 15;                        // A/B lane row / column
  const int half = lane >> 4;
  float* myLds = &ldsf0[wave * 16 * 36];
  float* myF1 = &ldsf1[wave * 16 * 68];
  unsigned* myV = &ldsv[wave * 16];

  // ---- async-stage W1 (64x32 f32) into LDS: 512 x b128 chunks ----
  {
    unsigned lds0 = (unsigned)(size_t)(&w1lds[threadIdx.x * 4]);
    unsigned lds1 = (unsigned)(size_t)(&w1lds[1024 + threadIdx.x * 4]);
    const float* g0p = W1 + threadIdx.x * 4;
    const float* g1p = W1 + 1024 + threadIdx.x * 4;
    asm volatile("global_load_async_to_lds_b128 %0, %1, off"
                 :: "v"(lds0), "v"(g0p) : "memory");
    asm volatile("global_load_async_to_lds_b128 %0, %1, off"
                 :: "v"(lds1), "v"(g1p) : "memory");
    asm volatile("s_wait_asynccnt 0x0" ::: "memory");
  }
  __syncthreads();

  // preload B operands (lane-resident, loop invariant)
  v2f Bf0[2][2];                                     // [kb][ntile], K = 4*kb + e + 2*half
#pragma unroll
  for (int kb = 0; kb < 2; ++kb)
#pragma unroll
    for (int nt = 0; nt < 2; ++nt)
#pragma unroll
      for (int e = 0; e < 2; ++e)
        Bf0[kb][nt][e] = Wf[(mrow + 16 * nt) * 8 + (4 * kb + e + 2 * half)];
  v2f B1m[8][4];
#pragma unroll
  for (int kb = 0; kb < 8; ++kb)
#pragma unroll
    for (int nt = 0; nt < 4; ++nt)
#pragma unroll
      for (int e = 0; e < 2; ++e)
        B1m[kb][nt][e] = w1lds[(mrow + 16 * nt) * 32 + (4 * kb + e + 2 * half)];
  float bias0[2] = {bf[mrow], bf[mrow + 16]};
  float bias1[4] = {b1[mrow], b1[mrow + 16], b1[mrow + 32], b1[mrow + 48]};

  const unsigned nwaves = gridDim.x * (blockDim.x >> 5);
  const unsigned gwave = blockIdx.x * (blockDim.x >> 5) + wave;
  const unsigned ntiles = NPTS / 16;                 // exactly divisible

  for (unsigned t = gwave; t < ntiles; t += nwaves) {
    const int pb = t * 16;
    const int pi = pb + mrow;
    __builtin_prefetch(feat + (size_t)(pb + 16) * 4, 0, 0);
    float4 f = ((const float4*)feat)[pi];
    int4 cc = ((const int4*)coors)[pi];
    unsigned key = (unsigned)cc.y * NAZ + (unsigned)cc.z;
    unsigned vid = rank[key];
    float4 vm = vacc[key];
    myV[mrow] = vid;                                 // both halves write same value
    float dist = __builtin_amdgcn_sqrtf(f.x * f.x + f.y * f.y + f.z * f.z);
    float fc0 = f.x - vm.x, fc1 = f.y - vm.y, fc2 = f.z - vm.z;

    // ---- layer 0: (16x8) @ (8x32) via 4 WMMAs ----
    // A element K = e + 2*half (+4 for a1): one v_cndmask per element
    v2f a0, a1;
    a0[0] = half ? f.z : f.x;
    a0[1] = half ? f.w : f.y;
    a1[0] = half ? fc1 : dist;
    a1[1] = half ? fc2 : fc0;
    v8f c0, c1;
#pragma unroll
    for (int r = 0; r < 8; ++r) { c0[r] = bias0[0]; c1[r] = bias0[1]; }
    c0 = wmma4(a0, Bf0[0][0], c0);
    c0 = wmma4(a1, Bf0[1][0], c0);
    c1 = wmma4(a0, Bf0[0][1], c1);
    c1 = wmma4(a1, Bf0[1][1], c1);
#pragma unroll
    for (int r = 0; r < 8; ++r) { c0[r] = fmaxf(c0[r], 0.f); c1[r] = fmaxf(c1[r], 0.f); }

    // ---- stage f0 tile to LDS (D-layout -> row-major) ----
#pragma unroll
    for (int r = 0; r < 8; ++r) {
      int row = r + 8 * half;
      myLds[row * 36 + mrow] = c0[r];
      myLds[row * 36 + mrow + 16] = c1[r];
    }
    asm volatile("s_wait_dscnt 0" ::: "memory");     // cross-lane LDS visibility

    // ---- layer 1: (16x32) @ (32x64) via 32 WMMAs ----
    v8f d0, d1, d2, d3;
#pragma unroll
    for (int r = 0; r < 8; ++r) {
      d0[r] = bias1[0]; d1[r] = bias1[1]; d2[r] = bias1[2]; d3[r] = bias1[3];
    }
#pragma unroll
    for (int kb = 0; kb < 8; ++kb) {
      v2f a = *(const v2f*)&myLds[mrow * 36 + 4 * kb + 2 * half];
      d0 = wmma4(a, B1m[kb][0], d0);
      d1 = wmma4(a, B1m[kb][1], d1);
      d2 = wmma4(a, B1m[kb][2], d2);
      d3 = wmma4(a, B1m[kb][3], d3);
    }

    // ---- stage f1 tile to LDS ----
#pragma unroll
    for (int r = 0; r < 8; ++r) {
      int row = r + 8 * half;
      myF1[row * 68 + mrow]      = d0[r];
      myF1[row * 68 + mrow + 16] = d1[r];
      myF1[row * 68 + mrow + 32] = d2[r];
      myF1[row * 68 + mrow + 48] = d3[r];
    }
    asm volatile("s_wait_dscnt 0" ::: "memory");

    // ---- coalesced b128 global stores for f0 (4/lane) and f1 (8/lane) ----
#pragma unroll
    for (int j = 0; j < 4; ++j) {
      int c = lane + 32 * j;                 // 0..127 chunks of 4 floats
      int row = c >> 3, ch = (c & 7) * 4;
      float4 vls = *(const float4*)&myLds[row * 36 + ch];
      *(float4*)&f0o[(size_t)(pb + row) * 32 + ch] = vls;
    }
#pragma unroll
    for (int j = 0; j < 8; ++j) {
      int c = lane + 32 * j;                 // 0..255 chunks of 4 floats
      int row = c >> 4, ch = (c & 15) * 4;
      float4 vls = *(const float4*)&myF1[row * 68 + ch];
      *(float4*)&f1o[(size_t)(pb + row) * 64 + ch] = vls;
    }

    // ---- segment-max via munged atomicMax (D-layout registers) ----
#pragma unroll
    for (int r = 0; r < 8; ++r) {
      int row = r + 8 * half;
      unsigned vv = myV[row];
      unsigned* vrow = vmaxm + (size_t)vv * 64;
      atomicMax(vrow + mrow,      mungf(d0[r]));
      atomicMax(vrow + mrow + 16, mungf(d1[r]));
      atomicMax(vrow + mrow + 32, mungf(d2[r]));
      atomicMax(vrow + mrow + 48, mungf(d3[r]));
    }
  }
}

// ---------------- K7: voxel compression (64->64) + relu -------------------
__global__ void __launch_bounds__(256) ffe_compress(
    const float* __restrict__ Wc, const float* __restrict__ bc,
    float* __restrict__ out, unsigned char* ws) {
  unsigned V = *(const unsigned*)(ws + V_OFF);
  unsigned tiles = (V + 15) >> 4;
  const unsigned gwave = blockIdx.x * (blockDim.x >> 5) + (threadIdx.x >> 5);
  if (gwave >= tiles) return;                        // wave-uniform
  const int lane = threadIdx.x & 31;
  const int mrow = lane & 15;
  const int half = lane >> 4;
  const unsigned* vmaxm = (const unsigned*)(ws + VMAXM_OFF);

  v8f acc[4];
#pragma unroll
  for (int nt = 0; nt < 4; ++nt) {
    float b = bc[mrow + 16 * nt];
#pragma unroll
    for (int r = 0; r < 8; ++r) acc[nt][r] = b;
  }
  const size_t vrow = (size_t)(gwave * 16 + mrow) * 64;
#pragma unroll
  for (int kb = 0; kb < 16; ++kb) {
    v2f a;
    a[0] = unmungf(vmaxm[vrow + 4 * kb + 2 * half]);
    a[1] = unmungf(vmaxm[vrow + 4 * kb + 2 * half + 1]);
#pragma unroll
    for (int nt = 0; nt < 4; ++nt) {
      v2f b;
      b[0] = Wc[(mrow + 16 * nt) * 64 + (4 * kb + 2 * half)];
      b[1] = Wc[(mrow + 16 * nt) * 64 + (4 * kb + 2 * half + 1)];
      acc[nt] = wmma4(a, b, acc[nt]);
    }
  }
#pragma unroll
  for (int r = 0; r < 8; ++r) {
    unsigned row = gwave * 16 + r + 8 * half;
    if (row < V) {
#pragma unroll
      for (int nt = 0; nt < 4; ++nt)
        out[(size_t)row * 64 + mrow + 16 * nt] = fmaxf(acc[nt][r], 0.f);
    }
  }
}

// ---------------- launch --------------------------------------------------
extern "C" void kernel_launch(void* const* d_in, const int* in_sizes, int n_in,
                              void* d_out, int out_size, void* d_ws, size_t ws_size,
                              hipStream_t stream) {
  (void)in_sizes; (void)n_in; (void)out_size; (void)ws_size;
  const float* feat  = (const float*)d_in[0];
  const int*   coors = (const int*)d_in[1];
  const float* pre_g = (const float*)d_in[2];
  const float* pre_b = (const float*)d_in[3];
  const float* W0    = (const float*)d_in[4];
  const float* g0    = (const float*)d_in[5];
  const float* b0    = (const float*)d_in[6];
  const float* W1    = (const float*)d_in[7];
  const float* b1    = (const float*)d_in[8];
  const float* Wc    = (const float*)d_in[9];
  const float* bc    = (const float*)d_in[10];
  float* out = (float*)d_out;
  unsigned char* ws = (unsigned char*)d_ws;

  ffe_init<<<(NKEY * 64) / 256, 256, 0, stream>>>(ws);
  ffe_scatter<<<(NPTS + 255) / 256, 256, 0, stream>>>(feat, coors, ws);
  ffe_scan<<<1, 1024, 0, stream>>>(out, ws);
  ffe_stats<<<1024, 256, 0, stream>>>(feat, coors, ws);
  ffe_fold<<<1, 32, 0, stream>>>(pre_g, pre_b, W0, g0, b0, ws);
  ffe_main<<<4096, 256, 0, stream>>>(feat, coors, W1, b1, out, ws);
  ffe_compress<<<256, 256, 0, stream>>>(Wc, bc, out, ws);
}